// ProbMPS_11261404250129
// MI455X (gfx1250) — compile-verified
//
#include <hip/hip_runtime.h>
#include <hip/hip_bf16.h>
#include <math.h>

// Problem constants (from reference)
#define SEQ_LEN   256
#define INPUT_DIM 64
#define BOND_DIM  128
#define BATCH     256
#define DD        (BOND_DIM * BOND_DIM)        // 16384 floats per matrix

typedef float v2f __attribute__((ext_vector_type(2)));
typedef float v8f __attribute__((ext_vector_type(8)));
typedef unsigned int u32x4 __attribute__((ext_vector_type(4)));
typedef int i32x8 __attribute__((ext_vector_type(8)));
typedef int i32x4 __attribute__((ext_vector_type(4)));

// -------- workspace layout (floats) --------
#define WS_RHOA   0
#define WS_RHOB   16384
#define WS_TRACES 32768
#define WS_PSILOG 33024

// -------- LDS layout for the step kernel (floats) --------
// rho_s   : 128x128            = 16384
// Abuf[2] : 128 rows x 132 pad = 16896 each (bank-conflict-free columns)
// Gs      : 16 rows x 132 pad  = 2112
#define AST       132                          // padded row stride for A / G
#define ABSZ      (BOND_DIM * AST)             // 16896 floats per A buffer
#define LDS_RHO   0
#define LDS_AB0   16384
#define LDS_G     (16384 + 2 * ABSZ)
#define LDS_FLOATS (16384 + 2 * ABSZ + 16 * AST)
#define LDS_BYTES  (LDS_FLOATS * 4)            // 209152 B < 320 KB WGP LDS

// ---------------- CDNA5 Tensor Data Mover (TENSORcnt) ----------------
#if defined(__has_builtin)
#if __has_builtin(__builtin_amdgcn_tensor_load_to_lds)
#define HAVE_TDM 1
#endif
#endif

__device__ __forceinline__ unsigned lds_byte_addr(const void* p) {
    return (unsigned)(size_t)(__attribute__((address_space(3))) const void*)p;
}

#ifdef HAVE_TDM
// Issue one TDM descriptor: copy a 128x128 f32 tile (row stride 128) from
// global into LDS; optional padding of 4 DWORDs after every 128 DWORDs
// (pad_interval=6 -> 2^(6+1)=128 DWORDs, pad_amount=3 -> 4 DWORDs) so rows
// land with the bank-conflict-free AST=132 stride.
__device__ __forceinline__ void tdm_load_tile(const float* gsrc, unsigned ldsb, int pad) {
    const unsigned long long ga = (unsigned long long)(size_t)gsrc;
    u32x4 g0;
    g0.x = 1u;                                            // count=1, load, user D#
    g0.y = ldsb;                                          // lds_addr (bytes)
    g0.z = (unsigned)(ga & 0xFFFFFFFFull);                // global_addr[31:0]
    g0.w = (unsigned)((ga >> 32) & 0x01FFFFFFull)         // global_addr[56:32]
         | (2u << 30);                                    // type=2 ("image")
    i32x8 g1;
    unsigned w0 = (2u << 16);                             // data_size = 4 bytes
    if (pad) w0 |= (1u << 20) | (6u << 22) | (3u << 25);  // pad_en, interval, amount
    g1[0] = (int)w0;
    g1[1] = (int)(128u << 16);     // tensor_dim0[15:0] in bits[63:48]
    g1[2] = (int)(128u << 16);     // tensor_dim0 hi=0 | tensor_dim1[15:0]
    g1[3] = (int)(128u << 16);     // tensor_dim1 hi=0 | tile_dim0 = 128
    g1[4] = 128;                   // tile_dim1 = 128, tile_dim2 = 0
    g1[5] = 128;                   // tensor_dim0_stride[31:0] = 128 elements
    g1[6] = 0;                     // stride hi | tensor_dim1_stride lo (unused, 2D)
    g1[7] = 0;
    const i32x4 z4 = {0, 0, 0, 0};               // groups 2/3 unused for 2D tiles
    const i32x8 z8 = {0, 0, 0, 0, 0, 0, 0, 0};   // extra group (6-arg toolchain)
    __builtin_amdgcn_tensor_load_to_lds(g0, g1, z4, z4, z8, 0);
}
#endif

// stage a 128x128 tile into LDS (TDM on wave 0, else cooperative copy)
__device__ __forceinline__ void stage_tile(const float* __restrict__ g,
                                           float* __restrict__ l,
                                           int pad, int tid) {
#ifdef HAVE_TDM
    if (tid < 32) tdm_load_tile(g, lds_byte_addr(l), pad);
#else
    const int stride = pad ? AST : BOND_DIM;
    #pragma unroll
    for (int j = 0; j < 16; ++j) {
        const int c   = tid + 256 * j;       // float4 chunk id, 0..4095
        const int row = c >> 5;              // 32 chunks per 128-float row
        const int q   = c & 31;
        *(float4*)(l + row * stride + q * 4) = *(const float4*)(g + row * BOND_DIM + q * 4);
    }
#endif
}

// wait for outstanding tensor DMAs issued by this thread's wave
__device__ __forceinline__ void stage_wait(int tid) {
#ifdef HAVE_TDM
    if (tid < 32) {
#if __has_builtin(__builtin_amdgcn_s_wait_tensorcnt)
        __builtin_amdgcn_s_wait_tensorcnt(0);
#else
        asm volatile("s_wait_tensorcnt 0x0" ::: "memory");
#endif
    }
#endif
}

// ---------------------------------------------------------------------------
// init: rhoA = outer(alpha, alpha); traces = 0
// ---------------------------------------------------------------------------
__global__ void mps_init_kernel(const float* __restrict__ edge, float* __restrict__ ws) {
    int idx = blockIdx.x * blockDim.x + threadIdx.x;   // 0..16383
    const float* alpha = edge;
    if (idx < DD) {
        int r = idx >> 7, c = idx & 127;
        ws[WS_RHOA + idx] = alpha[r] * alpha[c];
    }
    if (idx < SEQ_LEN) ws[WS_TRACES + idx] = 0.0f;
}

// ---------------------------------------------------------------------------
// psi chain: one block per batch element, 128 threads; h ping-pong in LDS.
// global_prefetch_b8 pulls next step's gathered matrix while this one runs.
// ---------------------------------------------------------------------------
__global__ void mps_psi_kernel(const int* __restrict__ input_data,
                               const float* __restrict__ core,
                               const float* __restrict__ edge,
                               float* __restrict__ ws) {
    __shared__ float hbuf[2][BOND_DIM];
    __shared__ float red[BOND_DIM];
    const int tid = threadIdx.x;         // 0..127
    const int b   = blockIdx.x;

    hbuf[0][tid] = edge[tid];            // alpha
    __syncthreads();

    int cur = 0;
    for (int t = 0; t < SEQ_LEN; ++t) {
        const int x = input_data[t * BATCH + b];
        const float* A = core + ((size_t)t * INPUT_DIM + (size_t)x) * DD;

        if (t + 1 < SEQ_LEN) {
            const int xn = input_data[(t + 1) * BATCH + b];
            const float* An = core + ((size_t)(t + 1) * INPUT_DIM + (size_t)xn) * DD;
            __builtin_prefetch(An + (size_t)tid * BOND_DIM, 0, 1);
        }

        const float* h = hbuf[cur];
        const float4* row4 = (const float4*)(A + (size_t)tid * BOND_DIM);
        float acc = 0.0f;
        #pragma unroll 8
        for (int j = 0; j < BOND_DIM / 4; ++j) {
            float4 a4 = row4[j];
            acc += a4.x * h[4 * j + 0];
            acc += a4.y * h[4 * j + 1];
            acc += a4.z * h[4 * j + 2];
            acc += a4.w * h[4 * j + 3];
        }
        __syncthreads();
        hbuf[cur ^ 1][tid] = acc;
        cur ^= 1;
        __syncthreads();
    }

    red[tid] = hbuf[cur][tid] * edge[BOND_DIM + tid];
    __syncthreads();
    for (int s = 64; s > 0; s >>= 1) {
        if (tid < s) red[tid] += red[tid + s];
        __syncthreads();
    }
    if (tid == 0) ws[WS_PSILOG + b] = logf(fabsf(red[0]));
}

// ---------------------------------------------------------------------------
// one transfer-matrix step, fully LDS-resident, TDM double-buffered A_d:
//   rho_out = (1/tr_{t-1}) * sum_d A_d rho_in A_d^T ;  traces[t] = trace
// 8 blocks (one 16-row stripe of rho' each) x 256 threads (8 waves).
// per d:  phase A (WMMA): G_d = A_d[i0:i0+16,:] @ rho          (LDS->LDS)
//         phase B (WMMA): acc += G_d @ A_d[l0:l0+16,:]^T       (LDS only)
// ---------------------------------------------------------------------------
__global__ void mps_step_kernel(const float* __restrict__ core,
                                const float* __restrict__ rho_in,
                                float* __restrict__ rho_out,
                                float* __restrict__ traces,
                                int t) {
    extern __shared__ float lds[];
    float* rho_s  = lds + LDS_RHO;
    float* AbBase = lds + LDS_AB0;       // two ABSZ buffers, selected by d&1
    float* Gs     = lds + LDS_G;

    const int tid  = threadIdx.x;
    const int lane = tid & 31;
    const int w    = tid >> 5;           // wave id 0..7
    const int m    = lane & 15;          // row (A-op) / col (B-op, C/D) index
    const int kh   = lane >> 4;          // K-pair select
    const int i0   = blockIdx.x * 16;    // row stripe of rho'
    const int l0   = w * 16;             // this wave's column tile of rho'

    const float inv_tr = (t > 0) ? (1.0f / traces[t - 1]) : 1.0f;
    const float* At = core + (size_t)t * INPUT_DIM * DD;

    // ---- prologue: DMA rho (unpadded) and A_0 (padded) into LDS ----
    stage_tile(rho_in, rho_s, /*pad=*/0, tid);
    stage_tile(At, AbBase, /*pad=*/1, tid);
    stage_wait(tid);
    __syncthreads();

    v8f acc = {0.f, 0.f, 0.f, 0.f, 0.f, 0.f, 0.f, 0.f};

    for (int d = 0; d < INPUT_DIM; ++d) {
        const int cur = d & 1;
        float* Ac = AbBase + cur * ABSZ;

        // DMA next A into the other buffer (overlaps both WMMA phases)
        if (d + 1 < INPUT_DIM)
            stage_tile(At + (size_t)(d + 1) * DD, AbBase + (cur ^ 1) * ABSZ,
                       /*pad=*/1, tid);

        // ---- phase A: G_d tile = A_d rows i0..i0+15 @ rho ----
        v8f g = {0.f, 0.f, 0.f, 0.f, 0.f, 0.f, 0.f, 0.f};
        #pragma unroll 4
        for (int k0 = 0; k0 < BOND_DIM; k0 += 4) {
            const int kk = k0 + 2 * kh;
            v2f a, bb;
            a.x  = Ac[(i0 + m) * AST + kk];              // broadcast rows, LDS
            a.y  = Ac[(i0 + m) * AST + kk + 1];
            bb.x = rho_s[kk * BOND_DIM + l0 + m];
            bb.y = rho_s[(kk + 1) * BOND_DIM + l0 + m];
            g = __builtin_amdgcn_wmma_f32_16x16x4_f32(
                    false, a, false, bb, (short)0, g, false, false);
        }
        #pragma unroll
        for (int v = 0; v < 8; ++v)
            Gs[(v + 8 * kh) * AST + l0 + m] = g[v];
        __syncthreads();                                  // G_d ready

        // ---- phase B: acc += G_d @ A_d^T  (all operands in LDS) ----
        #pragma unroll 4
        for (int k0 = 0; k0 < BOND_DIM; k0 += 4) {
            const int kk = k0 + 2 * kh;
            v2f a, bb;
            a.x  = Gs[m * AST + kk];
            a.y  = Gs[m * AST + kk + 1];
            bb.x = Ac[(l0 + m) * AST + kk];               // B-op = A_d^T tile
            bb.y = Ac[(l0 + m) * AST + kk + 1];
            acc = __builtin_amdgcn_wmma_f32_16x16x4_f32(
                    false, a, false, bb, (short)0, acc, false, false);
        }

        stage_wait(tid);                                  // next A_d landed
        __syncthreads();                                  // G_d consumed by all
    }

    // deferred trace normalization (scalar commutes with the linear map)
    float trace_part = 0.0f;
    #pragma unroll
    for (int v = 0; v < 8; ++v) {
        const float val = acc[v] * inv_tr;
        const int gr = i0 + v + 8 * kh;
        const int gc = l0 + m;
        rho_out[gr * BOND_DIM + gc] = val;
        if (gr == gc) trace_part += val;
    }
    if ((int)blockIdx.x == w && trace_part != 0.0f)
        atomicAdd(&traces[t], trace_part);
}

// ---------------------------------------------------------------------------
// finalize: log_norm = sum_t log(tr_t) + log( (omega^T rho_fin omega)/tr_255 )
// out[b] = 2*log|psi_b| - log_norm
// ---------------------------------------------------------------------------
__global__ void mps_final_kernel(const float* __restrict__ edge,
                                 const float* __restrict__ rho_fin,
                                 const float* __restrict__ ws,
                                 float* __restrict__ out) {
    __shared__ float zred[BOND_DIM];
    __shared__ float lg[SEQ_LEN];
    __shared__ float log_norm_s;
    const int tid = threadIdx.x;                 // 0..255
    const float* omega  = edge + BOND_DIM;
    const float* traces = ws + WS_TRACES;
    const float* pslog  = ws + WS_PSILOG;

    if (tid < BOND_DIM) {
        float vr = 0.0f;
        for (int c = 0; c < BOND_DIM; ++c)
            vr += rho_fin[tid * BOND_DIM + c] * omega[c];
        zred[tid] = omega[tid] * vr;
    }
    lg[tid] = logf(traces[tid]);
    __syncthreads();

    if (tid == 0) {
        float z = 0.0f;
        for (int r = 0; r < BOND_DIM; ++r) z += zred[r];
        z /= traces[SEQ_LEN - 1];
        float la = 0.0f;
        for (int s = 0; s < SEQ_LEN; ++s) la += lg[s];
        log_norm_s = la + logf(z);
    }
    __syncthreads();

    out[tid] = 2.0f * pslog[tid] - log_norm_s;
}

// ---------------------------------------------------------------------------
extern "C" void kernel_launch(void* const* d_in, const int* in_sizes, int n_in,
                              void* d_out, int out_size, void* d_ws, size_t ws_size,
                              hipStream_t stream) {
    (void)in_sizes; (void)n_in; (void)out_size; (void)ws_size;
    const int*   input_data = (const int*)  d_in[0];   // (256, 256)
    const float* core       = (const float*)d_in[1];   // (256, 64, 128, 128)
    const float* edge       = (const float*)d_in[2];   // (2, 128)
    float* out = (float*)d_out;                        // (256,)
    float* ws  = (float*)d_ws;

    float* rhoA   = ws + WS_RHOA;
    float* rhoB   = ws + WS_RHOB;
    float* traces = ws + WS_TRACES;

    mps_init_kernel<<<DD / 256, 256, 0, stream>>>(edge, ws);
    mps_psi_kernel<<<BATCH, BOND_DIM, 0, stream>>>(input_data, core, edge, ws);

    for (int t = 0; t < SEQ_LEN; ++t) {
        float* rin  = (t & 1) ? rhoB : rhoA;
        float* rout = (t & 1) ? rhoA : rhoB;
        mps_step_kernel<<<BOND_DIM / 16, 256, LDS_BYTES, stream>>>(core, rin, rout, traces, t);
    }
    // t=255 (odd) wrote rhoA
    mps_final_kernel<<<1, SEQ_LEN, 0, stream>>>(edge, rhoA, ws, out);
}